// SAW_25993142075773
// MI455X (gfx1250) — compile-verified
//
#include <hip/hip_runtime.h>
#include <stdint.h>

namespace {
constexpr int kB      = 8;
constexpr int kC      = 1024;
constexpr int kHW     = 64 * 64;     // 4096 floats per plane (16 KB)
constexpr int kLens   = 16;
constexpr int kGroups = 64;          // C / LENS
constexpr int kPlanes = kB * kC;     // 8192 output planes
}

__device__ __forceinline__ float dev_sigmoid(float v) {
    return 1.0f / (1.0f + __expf(-v));
}

// out[b, g*16+s, h, w] = sigmoid(weights[s, idx[s,g]]) * x[b, idx[s,g], h, w]
//
// One block per (b, oc) plane. The 16KB source plane is staged into LDS with
// the CDNA5 async global->LDS DMA path (global_load_async_to_lds_b128, tracked
// by ASYNCcnt), then scaled and stored with b128 accesses.
__global__ __launch_bounds__(256) void SAW_gather_scale(
    const float* __restrict__ x,
    const float* __restrict__ weights,
    const int*   __restrict__ idx,
    float*       __restrict__ out)
{
    const int plane = (int)blockIdx.x;          // 0 .. 8191
    const int b     = plane >> 10;              // / 1024
    const int oc    = plane & (kC - 1);
    const int g     = oc >> 4;                  // group
    const int s     = oc & (kLens - 1);         // selected class

    // Uniform (per-block) scalar metadata.
    const int   ch    = idx[s * kGroups + g];
    const float scale = dev_sigmoid(weights[s * kC + ch]);

    const float* __restrict__ src = x   + (size_t)(b * kC + ch) * kHW;
    float*       __restrict__ dst = out + (size_t)plane * kHW;
    const int t = (int)threadIdx.x;             // 0 .. 255

    __shared__ float tile[kHW];                 // 16 KB

    // --- Async global -> LDS stage of the whole plane ------------------------
    // Per the ISA, the instruction's immediate offset is added to BOTH the
    // global address and the LDS address, so one address pair + four immediate
    // offsets covers the full plane: 256 lanes x 16B x 4 = 16 KB.
    {
        // Low 32 bits of a generic shared pointer are the LDS byte address
        // (the high bits only select the LDS aperture).
        unsigned lds_addr = (unsigned)(uintptr_t)(&tile[t * 4]);
        unsigned long long gaddr = (unsigned long long)(uintptr_t)(src + t * 4);

        asm volatile("global_load_async_to_lds_b128 %0, %1, off"
                     :: "v"(lds_addr), "v"(gaddr) : "memory");
        asm volatile("global_load_async_to_lds_b128 %0, %1, off offset:4096"
                     :: "v"(lds_addr), "v"(gaddr) : "memory");
        asm volatile("global_load_async_to_lds_b128 %0, %1, off offset:8192"
                     :: "v"(lds_addr), "v"(gaddr) : "memory");
        asm volatile("global_load_async_to_lds_b128 %0, %1, off offset:12288"
                     :: "v"(lds_addr), "v"(gaddr) : "memory");
    }

    // Wait for this wave's async DMAs, then make all waves' LDS writes visible.
    asm volatile("s_wait_asynccnt 0" ::: "memory");
    __syncthreads();

    // --- LDS -> scale -> global store (b128 all the way) ---------------------
#pragma unroll
    for (int k = 0; k < 4; ++k) {
        const float4 v = *(const float4*)(&tile[t * 4 + k * 1024]);
        float4 r;
        r.x = v.x * scale;
        r.y = v.y * scale;
        r.z = v.z * scale;
        r.w = v.w * scale;
        *(float4*)(dst + t * 4 + k * 1024) = r;
    }
}

extern "C" void kernel_launch(void* const* d_in, const int* in_sizes, int n_in,
                              void* d_out, int out_size, void* d_ws, size_t ws_size,
                              hipStream_t stream) {
    (void)in_sizes; (void)n_in; (void)out_size; (void)d_ws; (void)ws_size;
    const float* x       = (const float*)d_in[0];  // (8, 1024, 64, 64) f32
    const float* weights = (const float*)d_in[1];  // (19, 1024) f32
    const int*   idx     = (const int*)  d_in[2];  // (19, 64) i32
    float*       out     = (float*)d_out;          // (8, 1024, 64, 64) f32

    SAW_gather_scale<<<dim3(kPlanes), dim3(256), 0, stream>>>(x, weights, idx, out);
}